// Net_81604378624770
// MI455X (gfx1250) — compile-verified
//
#include <hip/hip_runtime.h>

typedef __attribute__((ext_vector_type(2))) float v2f;
typedef __attribute__((ext_vector_type(8))) float v8f;

#define DHID 128
#define APAD 4
#define ASTRIDE (DHID + APAD)

// -------- helpers --------
__device__ __forceinline__ void gatomic_add(float* p, float v) {
#if defined(__HIP_DEVICE_COMPILE__)
  unsafeAtomicAdd(p, v);   // native global_atomic_add_f32
#else
  atomicAdd(p, v);
#endif
}

// -------- zero fill --------
__global__ void zero_f4(float4* __restrict__ p, long n4) {
  long i = (long)blockIdx.x * blockDim.x + threadIdx.x;
  long stride = (long)gridDim.x * blockDim.x;
  float4 z = make_float4(0.f, 0.f, 0.f, 0.f);
  for (; i < n4; i += stride) p[i] = z;
}

// -------- edge scatter-add: agg[dst[e]] += feat[src[e]] --------
// one wave (32 lanes) per edge, float4 per lane (128 dims). Gather is
// coalesced (512B per edge row); feat + agg are L2-resident (51MB each).
__global__ void edge_agg(const float* __restrict__ feat,
                         const int* __restrict__ src,
                         const int* __restrict__ dst,
                         float* __restrict__ agg, int n_edges) {
  int t = blockIdx.x * blockDim.x + threadIdx.x;
  int e = t >> 5;
  if (e >= n_edges) return;
  int lane = t & 31;
  int s = src[e];
  int d = dst[e];
  float4 v = ((const float4*)(feat + (size_t)s * DHID))[lane];
  float* o = agg + (size_t)d * DHID + lane * 4;
  gatomic_add(o + 0, v.x);
  gatomic_add(o + 1, v.y);
  gatomic_add(o + 2, v.z);
  gatomic_add(o + 3, v.w);
}

// -------- WMMA GEMM: out = relu(A[M x 128] @ W[128 x 128] + b) --------
// Block = 256 threads = 8 waves. Block owns 16 rows (one M-tile); wave w
// owns N-tile [16w, 16w+16). K loop: 32 x V_WMMA_F32_16X16X4_F32.
// A fragment (16x4 f32): lanes 0-15 hold M=lane, K=kh..kh+1 with kh=0;
// lanes 16-31 hold K=2..3. B fragment (4x16): lane holds B[kh..kh+1][n].
// C/D (16x16 f32): VGPR r, lanes 0-15 -> M=r, lanes 16-31 -> M=r+8, N=lane%16.
// If POOL: skip the store, relu-reduce over rows into pooled[128] instead.
template <bool POOL>
__global__ void gemm16(const float* __restrict__ A, const float* __restrict__ W,
                       const float* __restrict__ bias, float* __restrict__ out) {
  __shared__ float As[16 * ASTRIDE];
  __shared__ float red[DHID];

  const int m0 = blockIdx.x * 16;
  const int wave = threadIdx.x >> 5;
  const int lane = threadIdx.x & 31;
  const int n0 = wave * 16;
  const int nc = n0 + (lane & 15);
  const int kh = (lane >> 4) * 2;  // 0 or 2

  if (POOL) {
    if (threadIdx.x < DHID) red[threadIdx.x] = 0.f;
  }

  // cooperative A-tile load: 16 rows x 128 f32, padded stride, conflict-free
  {
    const float4* Ag = (const float4*)(A + (size_t)m0 * DHID);
    #pragma unroll
    for (int i = threadIdx.x; i < 16 * (DHID / 4); i += 256) {
      int row = i >> 5;          // /32 float4 per row
      int c4 = i & 31;
      *(float4*)&As[row * ASTRIDE + c4 * 4] = Ag[row * 32 + c4];
    }
  }
  __syncthreads();

  const float* arow = &As[(lane & 15) * ASTRIDE + kh];
  const float* bptr = W + (size_t)kh * DHID + nc;

  v8f c = {};
  #pragma unroll 8
  for (int k = 0; k < DHID; k += 4) {
    v2f a, b;
    float2 av = *(const float2*)&arow[k];       // ds_load_b64
    a.x = av.x; a.y = av.y;
    b.x = bptr[(size_t)k * DHID];
    b.y = bptr[(size_t)(k + 1) * DHID];
    c = __builtin_amdgcn_wmma_f32_16x16x4_f32(false, a, false, b,
                                              (short)0, c, false, false);
  }

  const float bv = bias[nc];
  const int mb = m0 + (lane >> 4) * 8;

  if (POOL) {
    float s = 0.f;
    #pragma unroll
    for (int r = 0; r < 8; ++r) {
      float v = c[r] + bv;
      s += (v > 0.f) ? v : 0.f;
    }
    atomicAdd(&red[nc], s);  // ds_add_f32 (lanes l and l+16 share nc)
    __syncthreads();
    if (threadIdx.x < DHID) gatomic_add(out + threadIdx.x, red[threadIdx.x]);
  } else {
    #pragma unroll
    for (int r = 0; r < 8; ++r) {
      float v = c[r] + bv;
      out[(size_t)(mb + r) * DHID + nc] = (v > 0.f) ? v : 0.f;
    }
  }
}

// -------- final tiny MLP: out[j] = relu(pooled[0:128] . W3[:,j] + b3[j]) --------
__global__ void final_mlp(const float* __restrict__ pooled,
                          const float* __restrict__ W3,
                          const float* __restrict__ b3,
                          float* __restrict__ out, int dout) {
  __shared__ float p[DHID];
  if (threadIdx.x < DHID) p[threadIdx.x] = pooled[threadIdx.x];
  __syncthreads();
  int j = threadIdx.x;
  if (j < dout) {
    float acc = b3[j];
    #pragma unroll 8
    for (int k = 0; k < DHID; ++k)
      acc = fmaf(p[k], W3[(size_t)k * dout + j], acc);
    out[j] = (acc > 0.f) ? acc : 0.f;
  }
}

extern "C" void kernel_launch(void* const* d_in, const int* in_sizes, int n_in,
                              void* d_out, int out_size, void* d_ws, size_t ws_size,
                              hipStream_t stream) {
  const float* feat = (const float*)d_in[0];
  const int*   src  = (const int*)d_in[1];
  const int*   dst  = (const int*)d_in[2];
  const float* W1   = (const float*)d_in[3];
  const float* b1   = (const float*)d_in[4];
  const float* W2   = (const float*)d_in[5];
  const float* b2   = (const float*)d_in[6];
  const float* W3   = (const float*)d_in[7];
  const float* b3   = (const float*)d_in[8];
  float* out = (float*)d_out;

  const int n_nodes = in_sizes[0] / DHID;  // 100000 (multiple of 16)
  const int n_edges = in_sizes[1];         // 1600000
  const int dout    = out_size;            // 256

  const size_t nf = (size_t)n_nodes * DHID;
  float* agg    = (float*)d_ws;          // 51.2 MB, reused for both layers
  float* x1     = agg + nf;              // 51.2 MB
  float* pooled = x1 + nf;               // 512 B

  const int eblocks = (n_edges * 32 + 255) / 256;
  const int gblocks = n_nodes / 16;

  // layer 1
  zero_f4<<<2048, 256, 0, stream>>>((float4*)agg, (long)(nf / 4));
  edge_agg<<<eblocks, 256, 0, stream>>>(feat, src, dst, agg, n_edges);
  gemm16<false><<<gblocks, 256, 0, stream>>>(agg, W1, b1, x1);

  // layer 2 (+ fused sum-pool; x2 never materialized)
  zero_f4<<<2048, 256, 0, stream>>>((float4*)agg, (long)(nf / 4));
  zero_f4<<<1, 32, 0, stream>>>((float4*)pooled, 32);
  edge_agg<<<eblocks, 256, 0, stream>>>(x1, src, dst, agg, n_edges);
  gemm16<true><<<gblocks, 256, 0, stream>>>(agg, W2, b2, pooled);

  // readout
  final_mlp<<<1, 256, 0, stream>>>(pooled, W3, b3, out, dout);
}